// RegularizedFMNet_27264452395494
// MI455X (gfx1250) — compile-verified
//
#include <hip/hip_runtime.h>

// ---------------------------------------------------------------------------
// RegularizedFMNet on MI455X (gfx1250, wave32).
//   B=8, V=20000, C=256, K=128, LMBDA=100, GAMMA=0.5
// Phase 1: A = Ex*Fx, Bm = Ey*Fy            (fp32 WMMA 16x16x4, dominant cost)
// Phase 2: AAt = A*A^T, BAt = Bm*A^T        (fp32 WMMA 16x16x4)
// Phase 3: 1024x SPD 128x128 solves in LDS  (GE without pivoting)
// LDS staging uses GLOBAL_LOAD_ASYNC_TO_LDS_B128 (ASYNCcnt) when available.
// Workspace: A (1MB) | Bm (1MB) | AAt (0.5MB) | BAt (0.5MB) = 3MB of d_ws.
// ---------------------------------------------------------------------------

typedef __attribute__((ext_vector_type(2))) float v2f;
typedef __attribute__((ext_vector_type(8))) float v8f;
typedef __attribute__((ext_vector_type(4))) int v4i;

#define B_ 8
#define K_ 128
#define C_ 256
#define LMBDA_ 100.0f

#if defined(__has_builtin)
#if __has_builtin(__builtin_amdgcn_global_load_async_to_lds_b128) && \
    __has_builtin(__builtin_amdgcn_s_wait_asynccnt)
#define USE_ASYNC_LDS 1
#endif
#endif
#ifndef USE_ASYNC_LDS
#define USE_ASYNC_LDS 0
#endif

#if USE_ASYNC_LDS
// Builtin signature (from hipcc diagnostic): param 1 is
// 'int __vector(4) __device__ *' (AS1 v4i*); LDS side is AS3 v4i*.
typedef __attribute__((address_space(1))) v4i* gv4i_p;
typedef __attribute__((address_space(3))) v4i* lv4i_p;

// Per-lane async gather: copies 16 bytes from the global address g to the LDS address l, tracked by ASYNCcnt.
__device__ __forceinline__ void async_b128(const void* g, void* l) {
  __builtin_amdgcn_global_load_async_to_lds_b128(
      (gv4i_p)(unsigned long long)(uintptr_t)g,
      (lv4i_p)(unsigned)(uintptr_t)l, 0, 0);
}
#endif

__device__ __forceinline__ v8f wmma4(v2f a, v2f b, v8f c) {
  // (neg_a, A, neg_b, B, c_mod, C, reuse_a, reuse_b)
  return __builtin_amdgcn_wmma_f32_16x16x4_f32(false, a, false, b, (short)0, c,
                                               false, false);
}

// ---------------------------------------------------------------------------
// Phase 1: Out[k,c] = sum_v E[k,v] * F[v,c]   for one batch set.
// Grid: (K/64, C/64, B). Block: 128 threads = 4 waves (2x2 wave grid),
// each wave computes a 32x32 tile via 2x2 WMMA accumulators.
// ---------------------------------------------------------------------------
__global__ __launch_bounds__(128) void proj_kernel(
    const float* __restrict__ Eg,   // [B, K, V]
    const float* __restrict__ Fg,   // [B, V, C]
    float* __restrict__ Outg,       // [B, K, C]
    int V) {
  const int k0 = blockIdx.x * 64;
  const int c0 = blockIdx.y * 64;
  const int b  = blockIdx.z;

  const float* E = Eg + (size_t)b * K_ * V;
  const float* F = Fg + (size_t)b * V * C_;
  float* Out     = Outg + (size_t)b * K_ * C_;

  const int tid  = threadIdx.x;
  const int lane = tid & 31;
  const int wid  = tid >> 5;
  const int wm   = wid >> 1;      // 0..1  (rows of 32)
  const int wn   = wid & 1;       // 0..1  (cols of 32)
  const int m16  = lane & 15;
  const int half = lane >> 4;

  __shared__ float Es[64][33];    // E tile: 64 rows(k) x 32 cols(v), padded
  __shared__ float Fs[32][65];    // F tile: 32 rows(v) x 64 cols(c), padded

  v8f acc[2][2] = {};

  for (int v0 = 0; v0 < V; v0 += 32) {
    __syncthreads();
    const bool full = (v0 + 32 <= V);
#if USE_ASYNC_LDS
    if (full) {
      // E tile 64x32: 512 b128 slots; F tile 32x64: 512 b128 slots.
      for (int s = tid; s < 64 * 8; s += 128) {
        const int row = s >> 3, c4 = (s & 7) << 2;
        async_b128(E + (size_t)(k0 + row) * V + v0 + c4, &Es[row][c4]);
      }
      for (int s = tid; s < 32 * 16; s += 128) {
        const int row = s >> 4, c4 = (s & 15) << 2;
        async_b128(F + (size_t)(v0 + row) * C_ + c0 + c4, &Fs[row][c4]);
      }
      __builtin_amdgcn_s_wait_asynccnt(0);
    } else
#endif
    {
      for (int s = tid; s < 64 * 8; s += 128) {
        const int row = s >> 3, c4 = (s & 7) << 2;
        float4 val = {0.f, 0.f, 0.f, 0.f};
        if (v0 + c4 + 3 < V)
          val = *(const float4*)(E + (size_t)(k0 + row) * V + v0 + c4);
        Es[row][c4 + 0] = val.x; Es[row][c4 + 1] = val.y;
        Es[row][c4 + 2] = val.z; Es[row][c4 + 3] = val.w;
      }
      for (int s = tid; s < 32 * 16; s += 128) {
        const int row = s >> 4, c4 = (s & 15) << 2;
        float4 val = {0.f, 0.f, 0.f, 0.f};
        if (v0 + row < V)
          val = *(const float4*)(F + (size_t)(v0 + row) * C_ + c0 + c4);
        Fs[row][c4 + 0] = val.x; Fs[row][c4 + 1] = val.y;
        Fs[row][c4 + 2] = val.z; Fs[row][c4 + 3] = val.w;
      }
    }
    __syncthreads();

    // Speculative prefetch of the next chunk (global_prefetch_b8).
    if (v0 + 32 < V) {
      __builtin_prefetch(E + (size_t)(k0 + (tid & 63)) * V + v0 + 32, 0, 0);
      __builtin_prefetch(
          F + (size_t)(v0 + 32 + (tid & 31)) * C_ + c0 + ((tid >> 5) << 4), 0, 0);
    }

#pragma unroll
    for (int vs = 0; vs < 32; vs += 4) {
      v2f a[2], bf[2];
#pragma unroll
      for (int tm = 0; tm < 2; ++tm) {
        const int row = wm * 32 + tm * 16 + m16;
        a[tm].x = Es[row][vs + 2 * half];
        a[tm].y = Es[row][vs + 2 * half + 1];
      }
#pragma unroll
      for (int tn = 0; tn < 2; ++tn) {
        const int col = wn * 32 + tn * 16 + m16;
        bf[tn].x = Fs[vs + 2 * half][col];
        bf[tn].y = Fs[vs + 2 * half + 1][col];
      }
#pragma unroll
      for (int tm = 0; tm < 2; ++tm)
#pragma unroll
        for (int tn = 0; tn < 2; ++tn)
          acc[tm][tn] = wmma4(a[tm], bf[tn], acc[tm][tn]);
    }
  }

  // Epilogue: VGPR j holds rows (j, j+8) of the 16x16 tile per lane-half.
#pragma unroll
  for (int tm = 0; tm < 2; ++tm)
#pragma unroll
    for (int tn = 0; tn < 2; ++tn)
#pragma unroll
      for (int j = 0; j < 8; ++j) {
        const int kg = k0 + wm * 32 + tm * 16 + j + half * 8;
        const int cg = c0 + wn * 32 + tn * 16 + m16;
        Out[(size_t)kg * C_ + cg] = acc[tm][tn][j];
      }
}

// ---------------------------------------------------------------------------
// Phase 2: Out[k,j] = sum_c L[k,c] * R[j,c]  (i.e. L * R^T), K_xK_ output.
// Grid: B*2 workgroups (which=0 -> AAt=A*A^T, which=1 -> BAt=Bm*A^T).
// Block: 256 threads = 8 waves (4x2 wave grid), wave tile = 32x64.
// ---------------------------------------------------------------------------
__global__ __launch_bounds__(256) void gram_kernel(
    const float* __restrict__ Ain,   // [B, K, C]
    const float* __restrict__ Bin,   // [B, K, C]
    float* __restrict__ AAt,         // [B, K, K]
    float* __restrict__ BAt) {       // [B, K, K]
  const int b     = blockIdx.x >> 1;
  const int which = blockIdx.x & 1;
  const float* L = (which ? Bin : Ain) + (size_t)b * K_ * C_;
  const float* R = Ain + (size_t)b * K_ * C_;
  float* Out     = (which ? BAt : AAt) + (size_t)b * K_ * K_;

  const int tid  = threadIdx.x;
  const int lane = tid & 31;
  const int wid  = tid >> 5;
  const int wm   = wid >> 1;      // 0..3 (rows of 32)
  const int wn   = wid & 1;       // 0..1 (cols of 64)
  const int m16  = lane & 15;
  const int half = lane >> 4;

  __shared__ float Ls[128][33];   // left rows x c-chunk
  __shared__ float Rs[128][33];   // right rows x c-chunk (accessed transposed)

  v8f acc[2][4] = {};

  for (int cc = 0; cc < C_; cc += 32) {
    __syncthreads();
#if USE_ASYNC_LDS
    for (int s = tid; s < 128 * 8; s += 256) {
      const int row = s >> 3, c4 = (s & 7) << 2;
      async_b128(L + (size_t)row * C_ + cc + c4, &Ls[row][c4]);
      async_b128(R + (size_t)row * C_ + cc + c4, &Rs[row][c4]);
    }
    __builtin_amdgcn_s_wait_asynccnt(0);
#else
    for (int s = tid; s < 128 * 8; s += 256) {
      const int row = s >> 3, c4 = (s & 7) << 2;
      float4 lv = *(const float4*)(L + (size_t)row * C_ + cc + c4);
      Ls[row][c4 + 0] = lv.x; Ls[row][c4 + 1] = lv.y;
      Ls[row][c4 + 2] = lv.z; Ls[row][c4 + 3] = lv.w;
      float4 rv = *(const float4*)(R + (size_t)row * C_ + cc + c4);
      Rs[row][c4 + 0] = rv.x; Rs[row][c4 + 1] = rv.y;
      Rs[row][c4 + 2] = rv.z; Rs[row][c4 + 3] = rv.w;
    }
#endif
    __syncthreads();

#pragma unroll
    for (int vs = 0; vs < 32; vs += 4) {
      v2f a[2], bf[4];
#pragma unroll
      for (int tm = 0; tm < 2; ++tm) {
        const int row = wm * 32 + tm * 16 + m16;
        a[tm].x = Ls[row][vs + 2 * half];
        a[tm].y = Ls[row][vs + 2 * half + 1];
      }
#pragma unroll
      for (int tn = 0; tn < 4; ++tn) {
        const int col = wn * 64 + tn * 16 + m16;  // output column j
        bf[tn].x = Rs[col][vs + 2 * half];        // B[kk][n] = R[j][c]
        bf[tn].y = Rs[col][vs + 2 * half + 1];
      }
#pragma unroll
      for (int tm = 0; tm < 2; ++tm)
#pragma unroll
        for (int tn = 0; tn < 4; ++tn)
          acc[tm][tn] = wmma4(a[tm], bf[tn], acc[tm][tn]);
    }
  }

#pragma unroll
  for (int tm = 0; tm < 2; ++tm)
#pragma unroll
    for (int tn = 0; tn < 4; ++tn)
#pragma unroll
      for (int j = 0; j < 8; ++j) {
        const int kg = wm * 32 + tm * 16 + j + half * 8;
        const int jg = wn * 64 + tn * 16 + m16;
        Out[(size_t)kg * K_ + jg] = acc[tm][tn][j];
      }
}

// ---------------------------------------------------------------------------
// Phase 3: per (b,i): M = AAt[b] + LMBDA*diag(D[b,i,:]); solve M x = BAt[b,i].
// SPD -> Gaussian elimination without pivoting, one workgroup per (b,i).
// 128 threads (thread t owns row t); 66KB LDS matrix (fits 320KB/WGP).
// ---------------------------------------------------------------------------
__global__ __launch_bounds__(128) void solve_kernel(
    const float* __restrict__ AAt,   // [B, K, K]
    const float* __restrict__ BAt,   // [B, K, K]
    const float* __restrict__ evx,   // [B, K]
    const float* __restrict__ evy,   // [B, K]
    float* __restrict__ Out) {       // [B, K, K]
  const int b = blockIdx.x >> 7;
  const int i = blockIdx.x & 127;
  const int t = threadIdx.x;

  __shared__ float Ms[K_][K_ + 1];
  __shared__ float rhs[K_];
  __shared__ float xv[K_];
  __shared__ float red[K_];

  // Load row t of AAt[b].
  const float* src = AAt + (size_t)b * K_ * K_ + (size_t)t * K_;
#pragma unroll 4
  for (int c = 0; c < K_; c += 4) {
    float4 v = *(const float4*)(src + c);
    Ms[t][c + 0] = v.x; Ms[t][c + 1] = v.y;
    Ms[t][c + 2] = v.z; Ms[t][c + 3] = v.w;
  }
  rhs[t] = BAt[(size_t)b * K_ * K_ + (size_t)i * K_ + t];

  // scale = max(max(evals_x[b]), max(evals_y[b]))
  const float ex = evx[b * K_ + t];
  const float ey = evy[b * K_ + t];
  red[t] = fmaxf(ex, ey);
  __syncthreads();
  for (int off = 64; off > 0; off >>= 1) {
    if (t < off) red[t] = fmaxf(red[t], red[t + off]);
    __syncthreads();
  }
  const float scale = red[0];

  // D[b,i,t] with gamma = 0.5 (pow -> sqrt).
  const float g2   = sqrtf(evy[b * K_ + i] / scale);
  const float inv2 = 1.0f / (g2 * g2 + 1.0f);
  const float a2   = g2 * inv2;
  const float g1   = sqrtf(ex / scale);
  const float inv1 = 1.0f / (g1 * g1 + 1.0f);
  const float a1   = g1 * inv1;
  const float mre  = a2 - a1;
  const float mim  = inv2 - inv1;
  const float d    = mre * mre + mim * mim;

  Ms[t][t] += LMBDA_ * d;
  __syncthreads();

  // Forward elimination (no pivoting; M is SPD).
  for (int p = 0; p < K_ - 1; ++p) {
    if (t > p) {
      const float f = Ms[t][p] / Ms[p][p];
      for (int c = p; c < K_; ++c) Ms[t][c] -= f * Ms[p][c];
      rhs[t] -= f * rhs[p];
    }
    __syncthreads();
  }

  // Back substitution.
  for (int p = K_ - 1; p >= 0; --p) {
    if (t == p) xv[p] = rhs[p] / Ms[p][p];
    __syncthreads();
    if (t < p) rhs[t] -= Ms[t][p] * xv[p];
    __syncthreads();
  }

  Out[(size_t)b * K_ * K_ + (size_t)i * K_ + t] = xv[t];
}

// ---------------------------------------------------------------------------
extern "C" void kernel_launch(void* const* d_in, const int* in_sizes, int n_in,
                              void* d_out, int out_size, void* d_ws,
                              size_t ws_size, hipStream_t stream) {
  const float* feat_x  = (const float*)d_in[0];  // [B, Vx, C]
  const float* feat_y  = (const float*)d_in[1];  // [B, Vy, C]
  const float* evals_x = (const float*)d_in[2];  // [B, K]
  const float* evals_y = (const float*)d_in[3];  // [B, K]
  const float* evtx    = (const float*)d_in[4];  // [B, K, Vx]
  const float* evty    = (const float*)d_in[5];  // [B, K, Vy]

  const int Vx = in_sizes[0] / (B_ * C_);
  const int Vy = in_sizes[1] / (B_ * C_);

  float* ws  = (float*)d_ws;
  float* A   = ws;                               // [B, K, C]
  float* Bm  = A + (size_t)B_ * K_ * C_;         // [B, K, C]
  float* AAt = Bm + (size_t)B_ * K_ * C_;        // [B, K, K]
  float* BAt = AAt + (size_t)B_ * K_ * K_;       // [B, K, K]
  float* out = (float*)d_out;                    // [B, K, K]

  // Phase 1: spectral projections.
  dim3 pgrid(K_ / 64, C_ / 64, B_);
  proj_kernel<<<pgrid, 128, 0, stream>>>(evtx, feat_x, A, Vx);
  proj_kernel<<<pgrid, 128, 0, stream>>>(evty, feat_y, Bm, Vy);

  // Phase 2: Gram matrices.
  gram_kernel<<<dim3(B_ * 2), 256, 0, stream>>>(A, Bm, AAt, BAt);

  // Phase 3: 1024 regularized solves.
  solve_kernel<<<dim3(B_ * K_), 128, 0, stream>>>(AAt, BAt, evals_x, evals_y,
                                                  out);
}